// MinusSpan_44693429682758
// MI455X (gfx1250) — compile-verified
//
#include <hip/hip_runtime.h>

// MinusSpan: B=8, T=4096, H=512.
// in[0]: x (B,T,2H) f32 ; in[1]: span_idxs (B,T,2) i32 ; out: (B,T,4H) f32.
// Memory-bound gather kernel: one 128-thread block per (b,t), block-uniform
// indices, async global->LDS staging (gfx1250 path), NT b128 stores.

#define T_DIM 4096
#define H_DIM 512

typedef __attribute__((ext_vector_type(4))) float v4f;
typedef __attribute__((ext_vector_type(4))) int   v4i;
typedef __attribute__((address_space(1))) v4i gas_v4i;
typedef __attribute__((address_space(3))) v4i las_v4i;
typedef __attribute__((address_space(3))) void las_void;

__device__ __forceinline__ void async_ld_b128(const float* g, float* l) {
#if __has_builtin(__builtin_amdgcn_global_load_async_to_lds_b128)
  __builtin_amdgcn_global_load_async_to_lds_b128(
      (gas_v4i*)g, (las_v4i*)l, /*offset=*/0, /*cpol=*/0);
#else
  unsigned lds_off = (unsigned)(size_t)(las_void*)l;
  asm volatile("global_load_async_to_lds_b128 %0, %1, off"
               :: "v"(lds_off), "v"(g) : "memory");
#endif
}

__device__ __forceinline__ void wait_asynccnt0() {
#if __has_builtin(__builtin_amdgcn_s_wait_asynccnt)
  __builtin_amdgcn_s_wait_asynccnt(0);
#else
  asm volatile("s_wait_asynccnt 0" ::: "memory");
#endif
}

__global__ __launch_bounds__(128) void MinusSpan_kernel(
    const float* __restrict__ x, const int* __restrict__ idx,
    float* __restrict__ out) {
  // Staging: row0=f_pre, row1=f_end, row2=b_start, row3=b_post (2KB each).
  __shared__ float lds[4 * H_DIM];

  const int bt  = blockIdx.x;      // b*T + t
  const int tid = threadIdx.x;     // 0..127, owns floats [4*tid, 4*tid+3]

  // Block-uniform span indices (scalar control flow for whole block).
  const int i = idx[2 * bt + 0];
  const int j = idx[2 * bt + 1];

  const int  t    = bt & (T_DIM - 1);
  const long base = (long)(bt - t) * (2 * H_DIM);   // b * T * 2H

  const int r_fpre = (i > 0) ? (i - 1) : 0;                   // clamped
  const int r_bpos = (j + 1 < T_DIM) ? (j + 1) : (T_DIM - 1);

  const int lane4 = tid * 4;
  const float* g_fpre = x + base + (long)r_fpre * (2 * H_DIM)         + lane4;
  const float* g_fend = x + base + (long)j      * (2 * H_DIM)         + lane4;
  const float* g_bsta = x + base + (long)i      * (2 * H_DIM) + H_DIM + lane4;
  const float* g_bpos = x + base + (long)r_bpos * (2 * H_DIM) + H_DIM + lane4;

  float* l_fpre = &lds[0 * H_DIM + lane4];
  float* l_fend = &lds[1 * H_DIM + lane4];
  float* l_bsta = &lds[2 * H_DIM + lane4];
  float* l_bpos = &lds[3 * H_DIM + lane4];

  // Async gather of the 4 rows into LDS (tracked by ASYNCcnt).
  async_ld_b128(g_fpre, l_fpre);
  async_ld_b128(g_fend, l_fend);
  async_ld_b128(g_bsta, l_bsta);
  async_ld_b128(g_bpos, l_bpos);
  wait_asynccnt0();
  // No barrier needed: each lane reads back exactly the slices it loaded.

  v4f fpre = *(const v4f*)l_fpre;
  v4f fend = *(const v4f*)l_fend;
  v4f bsta = *(const v4f*)l_bsta;
  v4f bpos = *(const v4f*)l_bpos;

  const v4f z = 0.0f;
  if (i == 0)          fpre = z;   // f_pre zeroed at left edge
  if (j + 1 >= T_DIM)  bpos = z;   // b_post zeroed at right edge

  v4f o0 = fend - fpre;
  v4f o1 = bsta - bpos;
  v4f o2 = fpre;
  v4f o3 = bpos;
  if (i == 0 && j == 0) { o0 = z; o1 = z; o2 = z; o3 = z; }  // invalid span

  // Streaming output (never re-read): non-temporal to keep input in L2.
  float* ob = out + (long)bt * (4 * H_DIM) + lane4;
  __builtin_nontemporal_store(o0, (v4f*)(ob + 0 * H_DIM));
  __builtin_nontemporal_store(o1, (v4f*)(ob + 1 * H_DIM));
  __builtin_nontemporal_store(o2, (v4f*)(ob + 2 * H_DIM));
  __builtin_nontemporal_store(o3, (v4f*)(ob + 3 * H_DIM));
}

extern "C" void kernel_launch(void* const* d_in, const int* in_sizes, int n_in,
                              void* d_out, int out_size, void* d_ws, size_t ws_size,
                              hipStream_t stream) {
  const float* x   = (const float*)d_in[0];
  const int*   idx = (const int*)d_in[1];
  float*       out = (float*)d_out;

  const int n_bt = 8 * T_DIM;  // B*T blocks, one per (b,t)
  MinusSpan_kernel<<<dim3(n_bt), dim3(128), 0, stream>>>(x, idx, out);
}